// CrossViewAttention_84593675862504
// MI455X (gfx1250) — compile-verified
//
#include <hip/hip_runtime.h>
#include <hip/hip_bf16.h>

typedef __attribute__((ext_vector_type(2))) float v2f;
typedef __attribute__((ext_vector_type(8))) float v8f;

#define DIM 128
#define KPANEL 32               // K-width staged in LDS per pass
#define LDS_STRIDE 36           // 32 + 4 pad dwords -> conflict-free ds_load_b64

// -----------------------------------------------------------------------------
// GEMM: out[r, :] = X[r, :] @ W^T + bias (+ residual), D = 128.
// Block = 128 threads = 4 wave32s; each wave owns a 16-row strip of X and all
// 128 output columns (8 16x16 tiles, 64 f32 accum VGPRs). W is staged into LDS
// in 32-wide K panels (batched global_load_b128 burst -> ds_store_b128) and
// consumed with conflict-free ds_load_b64. V_WMMA_F32_16X16X4_F32 keeps full
// fp32 precision.
// -----------------------------------------------------------------------------
__global__ void gemm128_wmma_f32(const float* __restrict__ X,
                                 const float* __restrict__ W,     // [128,128] row-major
                                 const float* __restrict__ bias,  // [128]
                                 const float* __restrict__ residual, // [nrows,128] or nullptr
                                 float* __restrict__ out,
                                 int nrows) {
    __shared__ float sW[DIM * LDS_STRIDE];   // 128 rows x (32+4) dwords = 18.4 KB

    const int tid   = threadIdx.x;
    const int lane  = tid & 31;
    const int wave  = tid >> 5;
    const int nstrips = (nrows + 15) >> 4;
    const int strip   = blockIdx.x * (blockDim.x >> 5) + wave;
    const bool valid  = (strip < nstrips);
    const int strip_c = valid ? strip : (nstrips - 1);   // clamp: keep barriers uniform

    const int m     = lane & 15;    // A row / B,C col within tile
    const int khalf = lane >> 4;    // 0: K=k0,k0+1  1: K=k0+2,k0+3

    v8f acc[8];
#pragma unroll
    for (int t = 0; t < 8; ++t) acc[t] = (v8f){0.f,0.f,0.f,0.f,0.f,0.f,0.f,0.f};

    const float* __restrict__ arow = X + (long)(strip_c * 16 + m) * DIM;
    const float* __restrict__ wrow = W + (long)tid * DIM;   // thread tid stages row tid

    for (int kp = 0; kp < DIM; kp += KPANEL) {
        // ---- stage W[:, kp..kp+31] into LDS: one batched load burst (single
        //      s_wait_loadcnt), then a burst of ds_store_b128 ----
        {
            float* dst = sW + tid * LDS_STRIDE;
            const float* src = wrow + kp;
            float4 stg[KPANEL / 4];
#pragma unroll
            for (int u = 0; u < KPANEL / 4; ++u)
                stg[u] = *(const float4*)(src + u * 4);
#pragma unroll
            for (int u = 0; u < KPANEL / 4; ++u)
                *(float4*)(dst + u * 4) = stg[u];
        }
        __syncthreads();

        // ---- preload this panel's A fragments (one batched burst, one wait) ----
        v2f aV[KPANEL / 4];
#pragma unroll
        for (int u = 0; u < KPANEL / 4; ++u)
            aV[u] = *(const v2f*)(arow + kp + u * 4 + 2 * khalf);

        // ---- 8 k-steps x 8 col tiles; B from LDS, loads batched ahead of WMMAs ----
#pragma unroll
        for (int kk = 0; kk < KPANEL; kk += 4) {
            const v2f a = aV[kk >> 2];
            v2f b[8];
#pragma unroll
            for (int t = 0; t < 8; ++t)
                b[t] = *(const v2f*)(sW + (t * 16 + m) * LDS_STRIDE + kk + 2 * khalf);
#pragma unroll
            for (int t = 0; t < 8; ++t)
                acc[t] = __builtin_amdgcn_wmma_f32_16x16x4_f32(
                    false, a, false, b[t], (short)0, acc[t], false, false);
        }
        __syncthreads();
    }

    if (valid) {
        // C/D layout: VGPR v -> row M = v + 8*(lane>>4), col N = lane&15
        const int rbase = strip * 16 + khalf * 8;
#pragma unroll
        for (int t = 0; t < 8; ++t) {
            const int col = t * 16 + m;
            const float bv = bias[col];
#pragma unroll
            for (int v = 0; v < 8; ++v) {
                const long r = rbase + v;
                float val = acc[t][v] + bv;
                if (residual) val += residual[r * DIM + col];
                out[r * DIM + col] = val;
            }
        }
    }
}

// -----------------------------------------------------------------------------
// Zero / init workspace accumulators.
// -----------------------------------------------------------------------------
__global__ void init_ws(unsigned* __restrict__ segmax_ord,
                        float* __restrict__ denom,
                        float* __restrict__ agg, int n_nodes) {
    const long i = (long)blockIdx.x * blockDim.x + threadIdx.x;
    if (i < (long)n_nodes * DIM) agg[i] = 0.0f;
    if (i < n_nodes) { segmax_ord[i] = 0u; denom[i] = 0.0f; }
}

// float <-> order-preserving unsigned (monotone map for atomicMax on floats)
__device__ __forceinline__ unsigned f32_to_ord(float f) {
    unsigned b = __float_as_uint(f);
    return b ^ ((b & 0x80000000u) ? 0xFFFFFFFFu : 0x80000000u);
}
__device__ __forceinline__ float ord_to_f32(unsigned o) {
    unsigned b = (o & 0x80000000u) ? (o ^ 0x80000000u) : ~o;
    return __uint_as_float(b);
}

// -----------------------------------------------------------------------------
// Pass 1: per-edge score = scale * dot(Qn[q], Kn[kv]); atomic segment max.
// One wave32 per edge; float4 per lane = coalesced 512B row reads (L2-hot).
// -----------------------------------------------------------------------------
__global__ void edge_scores(const float* __restrict__ Qn, const float* __restrict__ Kn,
                            const int* __restrict__ qidx, const int* __restrict__ kvidx,
                            float* __restrict__ scores, unsigned* __restrict__ segmax_ord,
                            int n_edges, float scale) {
    const int lane = threadIdx.x & 31;
    const long e = (long)blockIdx.x * (blockDim.x >> 5) + (threadIdx.x >> 5);
    if (e >= n_edges) return;
    const int q  = qidx[e];
    const int kv = kvidx[e];
    const float4 a = *(const float4*)(Qn + (long)q  * DIM + lane * 4);
    const float4 b = *(const float4*)(Kn + (long)kv * DIM + lane * 4);
    float p = a.x * b.x + a.y * b.y + a.z * b.z + a.w * b.w;
#pragma unroll
    for (int off = 16; off > 0; off >>= 1) p += __shfl_xor(p, off, 32);
    if (lane == 0) {
        const float s = p * scale;
        scores[e] = s;
        atomicMax(segmax_ord + q, f32_to_ord(s));
    }
}

// -----------------------------------------------------------------------------
// Pass 2: ex = exp(score - segmax[q]); atomic segment sum into denom.
// -----------------------------------------------------------------------------
__global__ void edge_exp(const float* __restrict__ scores, const int* __restrict__ qidx,
                         const unsigned* __restrict__ segmax_ord,
                         float* __restrict__ ex, float* __restrict__ denom, int n_edges) {
    const long e = (long)blockIdx.x * blockDim.x + threadIdx.x;
    if (e >= n_edges) return;
    const int q = qidx[e];
    const float m = ord_to_f32(segmax_ord[q]);
    const float v = __expf(scores[e] - m);
    ex[e] = v;
    atomicAdd(denom + q, v);
}

// -----------------------------------------------------------------------------
// Pass 3: agg[q, :] += (ex/denom[q]) * Vn[kv, :]. One wave32 per edge,
// float4 per lane, global_atomic_add_f32 into L2-resident destination.
// -----------------------------------------------------------------------------
__global__ void edge_scatter(const float* __restrict__ Vn,
                             const int* __restrict__ qidx, const int* __restrict__ kvidx,
                             const float* __restrict__ ex, const float* __restrict__ denom,
                             float* __restrict__ agg, int n_edges) {
    const int lane = threadIdx.x & 31;
    const long e = (long)blockIdx.x * (blockDim.x >> 5) + (threadIdx.x >> 5);
    if (e >= n_edges) return;
    const int q  = qidx[e];
    const int kv = kvidx[e];
    const float attn = ex[e] / denom[q];
    const float4 v = *(const float4*)(Vn + (long)kv * DIM + lane * 4);
    float* dst = agg + (long)q * DIM + lane * 4;
    atomicAdd(dst + 0, attn * v.x);
    atomicAdd(dst + 1, attn * v.y);
    atomicAdd(dst + 2, attn * v.z);
    atomicAdd(dst + 3, attn * v.w);
}

// -----------------------------------------------------------------------------
// Host-side launch
// -----------------------------------------------------------------------------
extern "C" void kernel_launch(void* const* d_in, const int* in_sizes, int n_in,
                              void* d_out, int out_size, void* d_ws, size_t ws_size,
                              hipStream_t stream) {
    const float* query = (const float*)d_in[0];
    const float* keyv  = (const float*)d_in[1];
    const int*   eidx  = (const int*)d_in[2];
    const float* Wq = (const float*)d_in[3];
    const float* bq = (const float*)d_in[4];
    const float* Wk = (const float*)d_in[5];
    const float* bk = (const float*)d_in[6];
    const float* Wv = (const float*)d_in[7];
    const float* bv = (const float*)d_in[8];
    const float* Wo = (const float*)d_in[9];
    const float* bo = (const float*)d_in[10];

    const int N = in_sizes[0] / DIM;
    const int E = in_sizes[2] / 2;
    const int* qidx  = eidx;
    const int* kvidx = eidx + E;
    const float scale = 1.0f / sqrtf((float)DIM);

    // bump-allocate workspace (256B aligned)
    char* ws = (char*)d_ws;
    auto alloc = [&](size_t bytes) -> char* {
        char* p = ws;
        ws += (bytes + 255) & ~(size_t)255;
        return p;
    };
    float*    Qn     = (float*)alloc((size_t)N * DIM * sizeof(float));
    float*    Kn     = (float*)alloc((size_t)N * DIM * sizeof(float));
    float*    Vn     = (float*)alloc((size_t)N * DIM * sizeof(float));
    float*    agg    = (float*)alloc((size_t)N * DIM * sizeof(float));
    float*    scores = (float*)alloc((size_t)E * sizeof(float));
    float*    ex     = (float*)alloc((size_t)E * sizeof(float));
    float*    denom  = (float*)alloc((size_t)N * sizeof(float));
    unsigned* segmax = (unsigned*)alloc((size_t)N * sizeof(unsigned));

    // GEMM config: 128 threads = 4 wave32s, each wave -> one 16-row strip
    const int strips = (N + 15) / 16;
    const dim3 gblk(128);
    const dim3 ggrd((strips + 3) / 4);

    // 1-3. node projections (WMMA fp32, W staged through LDS)
    gemm128_wmma_f32<<<ggrd, gblk, 0, stream>>>(query, Wq, bq, nullptr, Qn, N);
    gemm128_wmma_f32<<<ggrd, gblk, 0, stream>>>(keyv,  Wk, bk, nullptr, Kn, N);
    gemm128_wmma_f32<<<ggrd, gblk, 0, stream>>>(keyv,  Wv, bv, nullptr, Vn, N);

    // 4. init accumulators
    {
        const long tot = (long)N * DIM;
        init_ws<<<(unsigned)((tot + 255) / 256), 256, 0, stream>>>(segmax, denom, agg, N);
    }

    // 5-7. edge passes
    const int waves_per_blk = 256 / 32;
    edge_scores<<<(E + waves_per_blk - 1) / waves_per_blk, 256, 0, stream>>>(
        Qn, Kn, qidx, kvidx, scores, segmax, E, scale);
    edge_exp<<<(E + 255) / 256, 256, 0, stream>>>(scores, qidx, segmax, ex, denom, E);
    edge_scatter<<<(E + waves_per_blk - 1) / waves_per_blk, 256, 0, stream>>>(
        Vn, qidx, kvidx, ex, denom, agg, E);

    // 8. output projection + residual (WMMA fp32, fused bias + residual)
    gemm128_wmma_f32<<<ggrd, gblk, 0, stream>>>(agg, Wo, bo, query, (float*)d_out, N);
}